// DKVMNHeadGroup_21320217657935
// MI455X (gfx1250) — compile-verified
//
#include <hip/hip_runtime.h>

typedef __attribute__((ext_vector_type(2))) float v2f;
typedef __attribute__((ext_vector_type(4))) float v4f;
typedef __attribute__((ext_vector_type(8))) float v8f;

#define BB 4096
#define MM 128
#define DD 128

// One block handles 16 batch rows.
//  Phase 1 (waves 0-7): WMMA fp32 GEMM -> erase/add signals (16x128 each) into LDS.
//          (waves 8-15): stage write_weight tile (16x128) into LDS.
//  Phase 2 (all 16 waves): stream the [16 x M x D] memory update with
//          non-temporal 128-bit loads/stores (HBM-bound part).
__global__ __launch_bounds__(512)
void dkvmn_write_kernel(const float* __restrict__ ctrl,
                        const float* __restrict__ mem,
                        const float* __restrict__ ww,
                        const float* __restrict__ eW,
                        const float* __restrict__ eb,
                        const float* __restrict__ aW,
                        const float* __restrict__ ab,
                        float* __restrict__ out)
{
    __shared__ __align__(16) float sE[16 * DD];   // sigmoid(erase) signals
    __shared__ __align__(16) float sA[16 * DD];   // tanh(add) signals
    __shared__ float sW[16 * MM];                 // write weights

    const int tid  = threadIdx.x;
    const int lane = tid & 31;
    const int wave = tid >> 5;
    const int row0 = blockIdx.x * 16;             // first batch row of this block

    if (wave < 8) {
        // ---- GEMM phase: this wave computes a 16x16 tile of both signals ----
        // S[i][j] = sum_k ctrl[i][k] * W[j][k]   (i.e. ctrl @ W.T)
        const int col0 = wave * 16;
        const int mrow = lane & 15;               // A: M = lane&15
        const int kb   = (lane >> 4) << 1;        // A/B: K base = 0 or 2
        const int n    = lane & 15;               // B: N = lane&15
        const int col  = col0 + n;

        const float* Ap  = ctrl + (size_t)(row0 + mrow) * DD + kb;
        const float* Bep = eW   + (size_t)col * DD + kb;   // row of W == column of W.T
        const float* Bap = aW   + (size_t)col * DD + kb;

        v8f acc_e = {};
        v8f acc_a = {};
        #pragma unroll 8
        for (int k0 = 0; k0 < DD; k0 += 4) {
            v2f a  = *(const v2f*)(Ap  + k0);
            v2f be = *(const v2f*)(Bep + k0);
            v2f ba = *(const v2f*)(Bap + k0);
            acc_e = __builtin_amdgcn_wmma_f32_16x16x4_f32(
                        false, a, false, be, (short)0, acc_e, false, false);
            acc_a = __builtin_amdgcn_wmma_f32_16x16x4_f32(
                        false, a, false, ba, (short)0, acc_a, false, false);
        }

        // Bias + activation, then park signals in LDS.
        const float bias_e = eb[col];
        const float bias_a = ab[col];
        const int rbase = (lane >> 4) << 3;       // C/D: lanes 16-31 hold M=8..15
        #pragma unroll
        for (int r = 0; r < 8; ++r) {
            const int row = rbase + r;
            const float xe = acc_e[r] + bias_e;
            const float xa = acc_a[r] + bias_a;
            sE[row * DD + col] = 1.0f / (1.0f + __expf(-xe));
            sA[row * DD + col] = tanhf(xa);
        }
    } else {
        // ---- Stage write_weight tile into LDS (2048 floats) ----
        const int t2 = tid - 256;
        const float* wsrc = ww + (size_t)row0 * MM;
        #pragma unroll
        for (int i = 0; i < 8; ++i)
            sW[t2 + i * 256] = wsrc[t2 + i * 256];
    }
    __syncthreads();

    // ---- Streaming update phase (HBM-bound) ----
    // Thread owns (batch row b, float4 column dvec); loops over all M slots.
    const int b    = tid >> 5;        // 0..15  (whole wave shares one b)
    const int dvec = tid & 31;        // 0..31  (128-bit chunk index along D)

    const v4f e4 = *(const v4f*)(sE + b * DD + dvec * 4);
    const v4f a4 = *(const v4f*)(sA + b * DD + dvec * 4);
    const float* wrow = sW + b * MM;

    const size_t base = ((size_t)(row0 + b) * MM) * DD + (size_t)dvec * 4;
    const v4f* msrc = (const v4f*)(mem + base);
    v4f*       mdst = (v4f*)(out + base);

    #pragma unroll 4
    for (int m = 0; m < MM; ++m) {
        const float w = wrow[m];                          // wave-uniform LDS broadcast
        v4f v = __builtin_nontemporal_load(msrc + (size_t)m * (DD / 4));
        v4f r = v * (1.0f - e4 * w) + a4 * w;
        __builtin_nontemporal_store(r, mdst + (size_t)m * (DD / 4));
    }
}

extern "C" void kernel_launch(void* const* d_in, const int* in_sizes, int n_in,
                              void* d_out, int out_size, void* d_ws, size_t ws_size,
                              hipStream_t stream) {
    (void)in_sizes; (void)n_in; (void)d_ws; (void)ws_size; (void)out_size;
    const float* ctrl = (const float*)d_in[0];   // control_input [B, D]
    const float* mem  = (const float*)d_in[1];   // memory        [B, M, D]
    const float* ww   = (const float*)d_in[2];   // write_weight  [B, M]
    const float* eW   = (const float*)d_in[3];   // erase_W       [D, D]
    const float* eb   = (const float*)d_in[4];   // erase_b       [D]
    const float* aW   = (const float*)d_in[5];   // add_W         [D, D]
    const float* ab   = (const float*)d_in[6];   // add_b         [D]
    float* out = (float*)d_out;                  // new_memory    [B, M, D]

    dim3 grid(BB / 16);   // 256 blocks
    dim3 block(512);      // 16 waves: 8 WMMA waves + 8 staging waves
    hipLaunchKernelGGL(dkvmn_write_kernel, grid, block, 0, stream,
                       ctrl, mem, ww, eW, eb, aW, ab, out);
}